// TensorTrainGaussian_23553600651434
// MI455X (gfx1250) — compile-verified
//
#include <hip/hip_runtime.h>
#include <hip/hip_bf16.h>

// ---------------------------------------------------------------------------
// TensorTrain Gaussian likelihood, N=131072, M=32, K=16, for MI455X (gfx1250).
//   res[b,a](x) = softmax_b(W)[b,a] * N(x; mu,sigma)[b,a]
//               = 2^( c0'[b,a] + c1[b,a]*x + c2[b,a]*x^2 )
// with c0' = alpha*mu^2 + log2(wfac)  (weight folded into the exponent).
// The quadratic is computed by V_WMMA_F32_16X16X4_F32:
//   A row n = [x_n, x_n^2, x_n^2, 1]  (K2/K3 arranged so the result is correct
//   under either half-wave B row mapping)
//   B       = preswizzled per-lane table {c1,c2 | 0,c0'},  C = inline 0
// VALU does only exp2 + the contraction FMA; v (running left-product, 16
// floats per sample) lives in 8 VGPRs in the WMMA C/D layout; ds_bpermute
// broadcasts component a=t for the per-sample contraction.
// ---------------------------------------------------------------------------

#define NN 131072
#define MM 32
#define KK 16

typedef __attribute__((ext_vector_type(2))) float v2f;
typedef __attribute__((ext_vector_type(8))) float v8f;

// ---------------------------------------------------------------------------
// Kernel 1: precompute z = softmax(Wk0) and the per-step B-operand table.
// coefB[i][a][lane] (float2):
//   lane = b      (0..15) : {c1[b,a], c2[b,a]}        (K0/K1 rows)
//   lane = 16 + b (16..31): {0,       c0'[b,a]}       (zero row, K3 row)
//   alpha = -0.5*log2(e)/sigma^2
//   c1 = -2*alpha*mu ; c2 = alpha
//   c0' = alpha*mu^2 + log2( softmax_b(W[i,:,a])[b] / (sigma*sqrt(2*pi)) )
// ---------------------------------------------------------------------------
__global__ __launch_bounds__(256) void tt_precompute_kernel(
    const float* __restrict__ Wk0, const float* __restrict__ W,
    const float* __restrict__ mu, const float* __restrict__ sigma,
    float2* __restrict__ coefB, float* __restrict__ z) {
  const int tid = blockIdx.x * blockDim.x + threadIdx.x;

  if (tid < KK) {  // softmax of Wk0 (16 values), redundantly per thread
    float mx = -3.4e38f;
    for (int k = 0; k < KK; ++k) mx = fmaxf(mx, Wk0[k]);
    float s = 0.f;
    for (int k = 0; k < KK; ++k) s += __expf(Wk0[k] - mx);
    z[tid] = __expf(Wk0[tid] - mx) / s;
  }

  if (tid >= MM * KK) return;
  const int i = tid >> 4;   // step
  const int a = tid & 15;   // column of the 16x16 matrix

  // softmax over b of W[i, b, a]
  float w[KK];
  float mx = -3.4e38f;
  for (int b = 0; b < KK; ++b) {
    w[b] = W[(i * KK + b) * KK + a];
    mx = fmaxf(mx, w[b]);
  }
  float s = 0.f;
  for (int b = 0; b < KK; ++b) { w[b] = __expf(w[b] - mx); s += w[b]; }
  const float inv_s = 1.f / s;

  const float LOG2E = 1.4426950408889634f;
  const float INV_SQRT2PI = 0.3989422804014327f;
  float2* dst = coefB + (i * KK + a) * 32;
  for (int b = 0; b < KK; ++b) {
    const float m  = mu[b * KK + a];
    const float sg = sigma[b * KK + a];
    const float alpha = -0.5f * LOG2E / (sg * sg);
    const float c1 = -2.f * alpha * m;
    const float c2 = alpha;
    const float wf = (w[b] * inv_s) * (INV_SQRT2PI / sg);  // > 0 always
    const float c0p = alpha * m * m + __log2f(wf);
    dst[b]      = make_float2(c1, c2);
    dst[16 + b] = make_float2(0.f, c0p);
  }
}

// ---------------------------------------------------------------------------
// Kernel 2: main evaluation. 256 threads = 8 waves; each wave owns 16 samples.
// ---------------------------------------------------------------------------
__global__ __launch_bounds__(256) void tt_main_kernel(
    const float* __restrict__ X, const float2* __restrict__ coefB,
    const float* __restrict__ z, float* __restrict__ out) {
  const int lane = threadIdx.x & 31;
  const int wid  = threadIdx.x >> 5;
  const int half = lane >> 4;
  const int l15  = lane & 15;
  const int base = (blockIdx.x * 8 + wid) * 16;   // first sample of this wave
  const int samp = base + l15;                    // A-matrix row = l15

  // v in WMMA C/D layout: v[r] on lane c holds component (c&15) of sample
  // base + r + 8*(c>>4).  Init: every component k gets z[k].
  float v[8];
  const float zc = z[l15];
#pragma unroll
  for (int r = 0; r < 8; ++r) v[r] = zc;

  const int bp_base = (lane & 16) << 2;  // bpermute byte addr: ((lane&16)|t)*4

  for (int i = 0; i < MM; ++i) {
    const float x  = X[samp * MM + i];
    const float x2 = x * x;

    // A (16x4): lanes 0-15 -> {x, x2} (K0,K1); lanes 16-31 -> {x2, 1} (K2,K3).
    // Paired with B = {c1, c2 | 0, c0'}, D = c0' + c1*x + c2*x2 under either
    // plausible half-wave B row mapping (x2 term via K1 or K2; c0' via K3*1).
    v2f A;
    A.x = half ? x2 : x;
    A.y = half ? 1.f : x2;

    float acc[8];
#pragma unroll
    for (int r = 0; r < 8; ++r) acc[r] = 0.f;

    const float2* cs = coefB + i * 512 + lane;      // per-lane B operand
    __builtin_prefetch(coefB + (i + 1) * 512 + lane, 0, 3);  // next step

#pragma unroll
    for (int t = 0; t < 16; ++t) {  // t = column a of the 16x16 matrix
      const float2 bc = cs[t * 32];  // preswizzled B operand, one b64 load
      v2f B;
      B.x = bc.x;
      B.y = bc.y;

      // D[n, b] = c0'[b,t] + c1[b,t]*x_n + c2[b,t]*x_n^2  (C = inline 0)
      v8f D = __builtin_amdgcn_wmma_f32_16x16x4_f32(
          false, A, false, B, (short)0,
          v8f{0.f, 0.f, 0.f, 0.f, 0.f, 0.f, 0.f, 0.f}, false, false);

#pragma unroll
      for (int r = 0; r < 8; ++r) {
        const float p = __builtin_amdgcn_exp2f(D[r]);  // res[b,t] (weight folded)
        // broadcast v[n, a=t]: source lane = (lane&16)|t
        const float vb = __int_as_float(__builtin_amdgcn_ds_bpermute(
            bp_base + t * 4, __float_as_int(v[r])));
        acc[r] = fmaf(p, vb, acc[r]);
      }
    }

#pragma unroll
    for (int r = 0; r < 8; ++r) v[r] = acc[r];
  }

  // likelihood[n] = sum_b v[n,b]: butterfly over the 16 lanes of each half.
#pragma unroll
  for (int r = 0; r < 8; ++r) {
    float s = v[r];
#pragma unroll
    for (int k = 1; k <= 8; k <<= 1) {
      s += __int_as_float(__builtin_amdgcn_ds_bpermute(
          (lane ^ k) << 2, __float_as_int(s)));
    }
    // natural log(lik + eps) via v_log_f32 (log2) * ln(2)
    const float ll =
        __builtin_amdgcn_logf(s + 2.220446049250313e-16f) * 0.6931471805599453f;
    if (l15 == r) out[base + r + 8 * half] = ll;
  }
}

// ---------------------------------------------------------------------------
// Inputs (setup_inputs order): X (N*M f32), Wk0 (K f32), W (M*K*K f32),
// mu (K*K f32), sigma (K*K f32).  Output: N f32 log-likelihoods.
// Workspace: B-operand table 32*16*32*8 B = 128 KB, then z (16 floats).
// ---------------------------------------------------------------------------
extern "C" void kernel_launch(void* const* d_in, const int* in_sizes, int n_in,
                              void* d_out, int out_size, void* d_ws,
                              size_t ws_size, hipStream_t stream) {
  const float* X     = (const float*)d_in[0];
  const float* Wk0   = (const float*)d_in[1];
  const float* W     = (const float*)d_in[2];
  const float* mu    = (const float*)d_in[3];
  const float* sigma = (const float*)d_in[4];
  float* out = (float*)d_out;

  float2* coefB = (float2*)d_ws;
  float*  z     = (float*)((char*)d_ws + (size_t)MM * 512 * sizeof(float2));

  tt_precompute_kernel<<<2, 256, 0, stream>>>(Wk0, W, mu, sigma, coefB, z);
  tt_main_kernel<<<NN / 128, 256, 0, stream>>>(X, coefB, z, out);
}